// PtConv_10505490006249
// MI455X (gfx1250) — compile-verified
//
#include <hip/hip_runtime.h>
#include <hip/hip_bf16.h>

// fp32 WMMA fragments (gfx1250): A/B 16x4 / 4x16 = 2 VGPRs, C/D 16x16 = 8 VGPRs
typedef __attribute__((ext_vector_type(2))) float v2f;
typedef __attribute__((ext_vector_type(8))) float v8f;

#define BATCH   4
#define NPTS    8192
#define KNB     16
#define CIN     64
#define COUT    64
#define NC      16
#define DIM     3
#define L1IN    48      // DIM*NC
#define L1OUT   32      // 2*NC
#define TP      16      // points per workgroup
#define NTHREADS 128
#define NWAVES  4
#define FSTRIDE 1028    // fmat row stride in floats (1024 + 4) -> conflict-free lane reads

static __device__ __forceinline__ v8f wmma_f32(v2f a, v2f b, v8f c) {
    // D = A(16x4,f32) * B(4x16,f32) + C(16x16,f32)
    return __builtin_amdgcn_wmma_f32_16x16x4_f32(false, a, false, b,
                                                 (short)0, c, false, false);
}

__global__ __launch_bounds__(NTHREADS)
void ptconv_wmma_kernel(const float* __restrict__ inp,      // [B,N,CIN]
                        const float* __restrict__ points,   // [B,N,3]
                        const float* __restrict__ next_pts, // [B,N,3]
                        const int*   __restrict__ indices,  // [B,N,K] (int32)
                        const float* __restrict__ weight,   // [CIN,NC,COUT] -> [1024,64]
                        const float* __restrict__ bias,     // [COUT]
                        const float* __restrict__ centers,  // [DIM,NC] -> [48]
                        const float* __restrict__ l1w, const float* __restrict__ l1b,
                        const float* __restrict__ l2w, const float* __restrict__ l2b,
                        const float* __restrict__ l3w, const float* __restrict__ l3b,
                        float* __restrict__ out)             // [B,N,COUT]
{
    __shared__ float s_l1w[L1IN * L1OUT];
    __shared__ float s_l1b[L1OUT];
    __shared__ float s_l2w[L1OUT * NC];
    __shared__ float s_l2b[NC];
    __shared__ float s_l3w[NC * NC];
    __shared__ float s_l3b[NC];
    __shared__ float s_cent[DIM * NC];
    __shared__ float s_bias[COUT];
    __shared__ float s_rel[TP * KNB][4];        // relative coords (padded)
    __shared__ int   s_idx[TP * KNB];           // gathered neighbor indices
    __shared__ float s_d3[TP * KNB][NC];        // MLP output per (point,k)
    __shared__ float s_fmat[TP][FSTRIDE];       // f = feats^T @ d3, flattened per point
    __shared__ float s_s1[NWAVES][16 * 33];     // per-wave L1 staging (16x32, pad 33)
    __shared__ float s_s2[NWAVES][16 * 17];     // per-wave L2 staging (16x16, pad 17)

    const int tid    = threadIdx.x;
    const int wave   = tid >> 5;
    const int lane   = tid & 31;
    const int laneM  = lane & 15;          // M/N index within 16
    const int hiHalf = lane >> 4;          // 0|1: selects K {0,1} vs {2,3} in A/B frags
    const int koffA  = hiHalf * 2;
    const int hi8    = hiHalf * 8;         // C/D row offset for this lane half

    const int tilesPerBatch = NPTS / TP;
    const int b    = blockIdx.x / tilesPerBatch;
    const int tile = blockIdx.x % tilesPerBatch;
    const int n0   = tile * TP;

    // ---------------- Phase A: params to LDS, gather geometry, prefetch weight --------
    for (int i = tid; i < L1IN * L1OUT; i += NTHREADS) s_l1w[i] = l1w[i];
    for (int i = tid; i < L1OUT * NC;  i += NTHREADS) s_l2w[i] = l2w[i];
    for (int i = tid; i < NC * NC;     i += NTHREADS) s_l3w[i] = l3w[i];
    if (tid < L1OUT)   s_l1b[tid]  = l1b[tid];
    if (tid < NC)      { s_l2b[tid] = l2b[tid]; s_l3b[tid] = l3b[tid]; }
    if (tid < DIM*NC)  s_cent[tid] = centers[tid];
    if (tid < COUT)    s_bias[tid] = bias[tid];

    // keep the 256KB weight matrix hot in L2 (global_prefetch_b8)
    for (int i = tid; i < CIN * NC; i += NTHREADS * 2)
        __builtin_prefetch(weight + (size_t)i * COUT, 0, 1);

    for (int r = tid; r < TP * KNB; r += NTHREADS) {
        const int p = r >> 4, k = r & 15;
        const int n = n0 + p;
        const int idx = indices[((size_t)b * NPTS + n) * KNB + k];
        s_idx[r] = idx;
        const float* pp = points   + ((size_t)b * NPTS + idx) * DIM;
        const float* qq = next_pts + ((size_t)b * NPTS + n  ) * DIM;
        s_rel[r][0] = pp[0] - qq[0];
        s_rel[r][1] = pp[1] - qq[1];
        s_rel[r][2] = pp[2] - qq[2];
    }
    __syncthreads();

    // ---------------- Phase B: MLP over 256 rows (64 rows / wave, 4 tiles of 16) ------
    float* s1 = s_s1[wave];
    float* s2 = s_s2[wave];
    for (int t = 0; t < 4; ++t) {
        const int row0 = wave * 64 + t * 16;
        const int rA   = row0 + laneM;     // A-matrix row owned by this lane

        // L1: [16 x 48] @ [48 x 32] + b1, ReLU   (2 N-tiles x 12 K-steps)
        for (int nt = 0; nt < 2; ++nt) {
            v8f acc;
            const float bb = s_l1b[nt * 16 + laneM];
            #pragma unroll
            for (int r = 0; r < 8; ++r) acc[r] = bb;
            #pragma unroll
            for (int k0 = 0; k0 < L1IN; k0 += 4) {
                const int kA = k0 + koffA;
                v2f a, bf;
                // d0(row, j) = rel[row][j>>4] - centers[j]
                a.x  = s_rel[rA][(kA    ) >> 4] - s_cent[kA    ];
                a.y  = s_rel[rA][(kA + 1) >> 4] - s_cent[kA + 1];
                bf.x = s_l1w[(kA    ) * L1OUT + nt * 16 + laneM];
                bf.y = s_l1w[(kA + 1) * L1OUT + nt * 16 + laneM];
                acc = wmma_f32(a, bf, acc);
            }
            #pragma unroll
            for (int r = 0; r < 8; ++r) {
                float v = acc[r]; v = v > 0.f ? v : 0.f;
                s1[(r + hi8) * 33 + nt * 16 + laneM] = v;   // ReLU + C->A relayout
            }
        }

        // L2: [16 x 32] @ [32 x 16] + b2, ReLU   (8 K-steps)
        {
            v8f acc;
            const float bb = s_l2b[laneM];
            #pragma unroll
            for (int r = 0; r < 8; ++r) acc[r] = bb;
            #pragma unroll
            for (int k0 = 0; k0 < L1OUT; k0 += 4) {
                const int kA = k0 + koffA;
                v2f a, bf;
                a.x  = s1[laneM * 33 + kA];
                a.y  = s1[laneM * 33 + kA + 1];
                bf.x = s_l2w[(kA    ) * NC + laneM];
                bf.y = s_l2w[(kA + 1) * NC + laneM];
                acc = wmma_f32(a, bf, acc);
            }
            #pragma unroll
            for (int r = 0; r < 8; ++r) {
                float v = acc[r]; v = v > 0.f ? v : 0.f;
                s2[(r + hi8) * 17 + laneM] = v;
            }
        }

        // L3: [16 x 16] @ [16 x 16] + b3, ReLU   (4 K-steps) -> s_d3
        {
            v8f acc;
            const float bb = s_l3b[laneM];
            #pragma unroll
            for (int r = 0; r < 8; ++r) acc[r] = bb;
            #pragma unroll
            for (int k0 = 0; k0 < NC; k0 += 4) {
                const int kA = k0 + koffA;
                v2f a, bf;
                a.x  = s2[laneM * 17 + kA];
                a.y  = s2[laneM * 17 + kA + 1];
                bf.x = s_l3w[(kA    ) * NC + laneM];
                bf.y = s_l3w[(kA + 1) * NC + laneM];
                acc = wmma_f32(a, bf, acc);
            }
            #pragma unroll
            for (int r = 0; r < 8; ++r) {
                float v = acc[r]; v = v > 0.f ? v : 0.f;
                s_d3[row0 + r + hi8][laneM] = v;
            }
        }
    }
    __syncthreads();

    // ---------------- Phase C: f = feats^T @ d3 per point (4 points / wave) -----------
    for (int pi = 0; pi < 4; ++pi) {
        const int p = wave * 4 + pi;
        for (int mt = 0; mt < 4; ++mt) {               // c-tile: rows mt*16..mt*16+15
            v8f acc = {0.f, 0.f, 0.f, 0.f, 0.f, 0.f, 0.f, 0.f};
            #pragma unroll
            for (int k0 = 0; k0 < KNB; k0 += 4) {
                const int kA = k0 + koffA;
                const int i0 = s_idx[p * 16 + kA];
                const int i1 = s_idx[p * 16 + kA + 1];
                v2f a, bf;
                // A[c,k] = inp[b, idx[k], c] — gather straight from L2-resident inp
                a.x  = inp[((size_t)b * NPTS + i0) * CIN + mt * 16 + laneM];
                a.y  = inp[((size_t)b * NPTS + i1) * CIN + mt * 16 + laneM];
                bf.x = s_d3[p * 16 + kA    ][laneM];
                bf.y = s_d3[p * 16 + kA + 1][laneM];
                acc = wmma_f32(a, bf, acc);
            }
            #pragma unroll
            for (int r = 0; r < 8; ++r)                // f flattened: (c*16 + m)
                s_fmat[p][(mt * 16 + r + hi8) * 16 + laneM] = acc[r];
        }
    }
    __syncthreads();

    // ---------------- Phase D: out = fmat[16,1024] @ W[1024,64] / K + bias ------------
    {
        const int ocol = wave * 16 + laneM;            // one 16-col N-tile per wave
        v8f acc = {0.f, 0.f, 0.f, 0.f, 0.f, 0.f, 0.f, 0.f};
        #pragma unroll 4
        for (int k0 = 0; k0 < CIN * NC; k0 += 4) {     // 256 K-steps
            const int kA = k0 + koffA;
            v2f a, bf;
            a.x  = s_fmat[laneM][kA];
            a.y  = s_fmat[laneM][kA + 1];
            bf.x = weight[(size_t)(kA    ) * COUT + ocol];
            bf.y = weight[(size_t)(kA + 1) * COUT + ocol];
            acc = wmma_f32(a, bf, acc);
        }
        const float bb = s_bias[ocol];
        #pragma unroll
        for (int r = 0; r < 8; ++r) {
            const int p = r + hi8;                     // D row = point index
            out[((size_t)b * NPTS + n0 + p) * COUT + ocol] =
                acc[r] * (1.0f / KNB) + bb;
        }
    }
}

extern "C" void kernel_launch(void* const* d_in, const int* in_sizes, int n_in,
                              void* d_out, int out_size, void* d_ws, size_t ws_size,
                              hipStream_t stream) {
    (void)in_sizes; (void)n_in; (void)out_size; (void)d_ws; (void)ws_size;
    const float* inp      = (const float*)d_in[0];
    const float* points   = (const float*)d_in[1];
    const float* next_pts = (const float*)d_in[2];
    const int*   indices  = (const int*)  d_in[3];   // default-JAX int32
    const float* weight   = (const float*)d_in[4];
    const float* bias     = (const float*)d_in[5];
    const float* centers  = (const float*)d_in[6];
    const float* l1w      = (const float*)d_in[7];
    const float* l1b      = (const float*)d_in[8];
    const float* l2w      = (const float*)d_in[9];
    const float* l2b      = (const float*)d_in[10];
    const float* l3w      = (const float*)d_in[11];
    const float* l3b      = (const float*)d_in[12];
    float* out = (float*)d_out;

    const int grid = BATCH * (NPTS / TP);   // 2048 workgroups, 16 points each
    ptconv_wmma_kernel<<<grid, NTHREADS, 0, stream>>>(
        inp, points, next_pts, indices, weight, bias, centers,
        l1w, l1b, l2w, l2b, l3w, l3b, out);
}